// FeastNet_37830071943476
// MI455X (gfx1250) — compile-verified
//
#include <hip/hip_runtime.h>

// FeastNet graph conv: B=2, V=50000, NB=16, C=64, K=8, O=64.
// All three contractions on V_WMMA_F32_16X16X4_F32 (fp32-exact):
//   1) logits  L[16n,8k]   = diff[16n,64c] @ u[64c,8k]       (16 WMMA / vertex)
//   2) S       S[64c,8k]   = patchT[64c,16n] @ q[16n,8k]     (16 WMMA / vertex)
//   3) output  Out[16v,64o]= S[16v,512ck] @ W[512ck,64o]     (512 WMMA / tile)
// Masking is applied to q rows (q[n]=0 when adj==0), so the gather needs no
// conditional data path -> it can run on the async load-to-LDS engine
// (GLOBAL_LOAD_ASYNC_TO_LDS_B32 / ASYNCcnt), double-buffered and overlapped
// with the WMMA work of the previous vertex.

#define B_  2
#define V_  50000
#define NB_ 16
#define C_  64
#define K_  8
#define O_  64
#define CK_ 512
#define TILES_PER_B 3125      // V_/16
#define PT_STRIDE   17        // odd -> conflict-free strided LDS access
#define SS_STRIDE   516       // 516 % 64 == 4 -> all 32 lanes on distinct banks

#if defined(__gfx1250__) && __has_builtin(__builtin_amdgcn_global_load_async_to_lds_b32)
#define USE_ASYNC 1
#else
#define USE_ASYNC 0
#endif

typedef __attribute__((ext_vector_type(2))) float v2f;
typedef __attribute__((ext_vector_type(8))) float v8f;

__launch_bounds__(32)
__global__ void feastnet_kernel(const float* __restrict__ x,
                                const float* __restrict__ u,
                                const float* __restrict__ W,
                                const int*   __restrict__ adj,
                                float* __restrict__ out)
{
    __shared__ float sh_x[16][C_];              // center rows            (4 KB)
    __shared__ float sh_S[16][SS_STRIDE];       // S, padded stride       (33 KB)
    __shared__ float sh_pT[2][C_][PT_STRIDE];   // transposed patches, x2 (8.5 KB)
    __shared__ float sh_q[16][16];              // q in B layout          (1 KB)
    __shared__ int   sh_adj[16][NB_];           //                        (1 KB)
    __shared__ float sh_u[C_][16];              // u zero-padded          (4 KB)
    __shared__ float sh_invdeg[16];
    __shared__ unsigned sh_mask[16];

    const int lane = threadIdx.x;            // 0..31 (wave32)
    const int t    = blockIdx.x;             // tile id, 0..6249
    const int b    = t / TILES_PER_B;
    const int v0   = (t % TILES_PER_B) * 16;
    const float* __restrict__ xb = x + (size_t)b * V_ * C_;

    // ---- stage adj first so the v=0 gather can issue ASAP ----
    for (int i = lane; i < 16 * NB_; i += 32)
        ((int*)sh_adj)[i] = adj[(size_t)(v0 + (i >> 4)) * NB_ + (i & 15)];
    __syncthreads();

    // gather 16 neighbor rows of vertex vv, transposed into pT[buf][c][n].
    // Row clamped to 0 for adj==0 (finite garbage; q row is zeroed instead).
    auto gather = [&](int vv, int buf) {
        for (int i = lane; i < NB_ * C_; i += 32) {
            const int n = i >> 6, c = i & 63;
            const int j = sh_adj[vv][n];
            const int row = (j > 0) ? (j - 1) : 0;
            const float* g = xb + (size_t)row * C_ + c;
#if USE_ASYNC
            __builtin_amdgcn_global_load_async_to_lds_b32(
                (__attribute__((address_space(1))) int*)g,
                (__attribute__((address_space(3))) int*)&sh_pT[buf][c][n],
                0, 0);
#else
            sh_pT[buf][c][n] = *g;
#endif
        }
    };

    gather(0, 0);   // kick off vertex 0 while we stage the rest

    // per-vertex valid-mask and 1/deg (lanes 0-15, once per tile)
    if (lane < 16) {
        unsigned mk = 0; int dg = 0;
        #pragma unroll
        for (int n = 0; n < NB_; ++n) {
            const int nz = (sh_adj[lane][n] != 0);
            mk |= (unsigned)nz << n;
            dg += nz;
        }
        sh_mask[lane]   = mk;
        sh_invdeg[lane] = (dg > 0) ? 1.0f / (float)dg : 0.0f;
    }
    // stage u (zero-padded to 16 cols) and center rows
    for (int i = lane; i < C_ * 16; i += 32) {
        const int c = i >> 4, k = i & 15;
        sh_u[c][k] = (k < K_) ? u[c * K_ + k] : 0.0f;
    }
    for (int i = lane; i < 16 * C_; i += 32)
        ((float*)sh_x)[i] = xb[(size_t)(v0 + (i >> 6)) * C_ + (i & 63)];
    __syncthreads();

    const int m  = lane & 15;           // A-row / B-col index within a 16-tile
    const int kh = (lane >> 4) * 2;     // K-pair selector for A/B operands
    const int mb = (lane < 16) ? 0 : 8; // D-layout row base

    // ================= Phase 1: attention + S, all on WMMA =================
    for (int v = 0; v < 16; ++v) {
        const int cur = v & 1;
#if USE_ASYNC
        // pT[cur] produced by the async engine; fence before consuming
  #if __has_builtin(__builtin_amdgcn_s_wait_asynccnt)
        __builtin_amdgcn_s_wait_asynccnt(0);
  #else
        asm volatile("s_wait_asynccnt 0x0" ::: "memory");
  #endif
#endif
        __syncthreads();
        if (v < 15) gather(v + 1, cur ^ 1);   // overlap next gather with WMMAs

        const unsigned msk = sh_mask[v];
        const float    idg = sh_invdeg[v];

        // ---- logits: L[16n,16k] = diff[16n,64c] @ u[64c,16k], 16 K-steps ----
        v8f la0 = {}, la1 = {};               // two chains for XDL ILP
        #pragma unroll
        for (int ks = 0; ks < 16; ks += 2) {
            {
                const int c = ks * 4 + kh;
                v2f a, bb;
                a.x  = sh_x[v][c]     - sh_pT[cur][c][m];
                a.y  = sh_x[v][c + 1] - sh_pT[cur][c + 1][m];
                bb.x = sh_u[c][m];
                bb.y = sh_u[c + 1][m];
                la0 = __builtin_amdgcn_wmma_f32_16x16x4_f32(false, a, false, bb, (short)0, la0, false, false);
            }
            {
                const int c = (ks + 1) * 4 + kh;
                v2f a, bb;
                a.x  = sh_x[v][c]     - sh_pT[cur][c][m];
                a.y  = sh_x[v][c + 1] - sh_pT[cur][c + 1][m];
                bb.x = sh_u[c][m];
                bb.y = sh_u[c + 1][m];
                la1 = __builtin_amdgcn_wmma_f32_16x16x4_f32(false, a, false, bb, (short)0, la1, false, false);
            }
        }

        // ---- softmax over k on D-layout (row n = mb+j, col k = m) ----
        #pragma unroll
        for (int j = 0; j < 8; ++j) {
            const float l = la0[j] + la1[j];
            float mx = l;
            mx = fmaxf(mx, __shfl_xor(mx, 1, 32));
            mx = fmaxf(mx, __shfl_xor(mx, 2, 32));
            mx = fmaxf(mx, __shfl_xor(mx, 4, 32));
            const float e = __expf(l - mx);
            float s = e;
            s += __shfl_xor(s, 1, 32);
            s += __shfl_xor(s, 2, 32);
            s += __shfl_xor(s, 4, 32);
            // zero masked neighbor rows and the k>=8 padding cols
            const bool valid = (m < K_) && ((msk >> (mb + j)) & 1u);
            sh_q[mb + j][m] = valid ? e * (idg / s) : 0.0f;
        }
        __syncthreads();

        // ---- S: S[64c,16k] = pT[64c,16n] @ q[16n,16k], 4 M-tiles x 4 K-steps
        v8f sacc[4] = {};
        #pragma unroll
        for (int ksn = 0; ksn < 4; ++ksn) {
            const int n0 = ksn * 4 + kh;
            v2f bb;
            bb.x = sh_q[n0][m];
            bb.y = sh_q[n0 + 1][m];
            #pragma unroll
            for (int mt = 0; mt < 4; ++mt) {
                v2f a;                          // consecutive -> ds_load_b64
                a.x = sh_pT[cur][mt * 16 + m][n0];
                a.y = sh_pT[cur][mt * 16 + m][n0 + 1];
                sacc[mt] = __builtin_amdgcn_wmma_f32_16x16x4_f32(false, a, false, bb, (short)0, sacc[mt], false, false);
            }
        }
        // store S (valid cols k<8): element [c = mt*16 + mb + j][k = m]
        if (m < K_) {
            #pragma unroll
            for (int mt = 0; mt < 4; ++mt)
                #pragma unroll
                for (int j = 0; j < 8; ++j)
                    sh_S[v][(mt * 16 + mb + j) * K_ + m] = sacc[mt][j];
        }
        __syncthreads();
    }

    // ============ Phase 2: Out[16,64] = S[16,512] @ W[512,64] ============
    v8f acc[4] = {};
    #pragma unroll 4
    for (int ks = 0; ks < CK_ / 4; ++ks) {
        const int k0 = ks * 4;
        v2f a;                                 // padded stride: conflict-free
        a.x = sh_S[m][k0 + kh];
        a.y = sh_S[m][k0 + kh + 1];
        const float* __restrict__ wr = W + (size_t)(k0 + kh) * O_ + m;  // L2-hot
        #pragma unroll
        for (int nt = 0; nt < 4; ++nt) {
            v2f bb;
            bb.x = wr[nt * 16];
            bb.y = wr[nt * 16 + O_];
            acc[nt] = __builtin_amdgcn_wmma_f32_16x16x4_f32(false, a, false, bb, (short)0, acc[nt], false, false);
        }
    }

    // ---- store D: lane<16 -> rows j, lane>=16 -> rows j+8; o = nt*16 + m
    float* __restrict__ ob = out + ((size_t)b * V_ + v0) * O_;
    #pragma unroll
    for (int nt = 0; nt < 4; ++nt) {
        const int o = nt * 16 + m;
        #pragma unroll
        for (int j = 0; j < 8; ++j)
            ob[(size_t)(mb + j) * O_ + o] = acc[nt][j];
    }
}

extern "C" void kernel_launch(void* const* d_in, const int* in_sizes, int n_in,
                              void* d_out, int out_size, void* d_ws, size_t ws_size,
                              hipStream_t stream) {
    const float* x   = (const float*)d_in[0];
    const float* u   = (const float*)d_in[1];
    const float* W   = (const float*)d_in[2];
    const int*   adj = (const int*)d_in[3];
    float*       out = (float*)d_out;

    dim3 grid(B_ * TILES_PER_B);   // 6250 tiles of 16 vertices
    dim3 block(32);                // one wave32 per tile
    feastnet_kernel<<<grid, block, 0, stream>>>(x, u, W, adj, out);
}